// RegionLoss_3702261809894
// MI455X (gfx1250) — compile-verified
//
#include <hip/hip_runtime.h>
#include <hip/hip_bf16.h>
#include <math.h>

typedef __attribute__((ext_vector_type(2))) float v2f;
typedef __attribute__((ext_vector_type(8))) float v8f;
typedef unsigned int uint32;

// Problem constants from setup_inputs()
constexpr int B = 16, A = 3, H = 76, W = 76, C = 80, T = 32;
constexpr int HW = H * W;
constexpr int NCELL = B * A * H * W;   // 277248  (= 1083 * 256 exactly)
constexpr int NGT = B * T;             // 512
constexpr float IGNORE_THRES = 0.5f;
constexpr float EPS_BCE = 1e-7f;

// accumulator slots
// 0..3: smoothL1 x,y,w,h   4: sum -log(pc) on mask   5: sum -log(1-pc) on conf_false
// 6: sum CE   7: nM count   8: nF count   9: n_correct (int alias)

__device__ __forceinline__ float sigmoidf(float x) { return 1.0f / (1.0f + __expf(-x)); }
__device__ __forceinline__ float smooth_l1(float p, float t) {
    float d = fabsf(p - t);
    return (d < 1.0f) ? 0.5f * d * d : d - 0.5f;
}

__global__ void k_init(uint32* cellGT, uint32* cellCZ, float* acc) {
    int i = blockIdx.x * blockDim.x + threadIdx.x;
    if (i < NCELL) { cellGT[i] = 0u; cellCZ[i] = 0u; }
    if (i < 16) acc[i] = 0.0f;
}

__global__ void k_build(const float* __restrict__ xr, const float* __restrict__ xc,
                        const float* __restrict__ tgt, const float* __restrict__ anc,
                        uint32* cellGT, uint32* cellCZ, float* acc) {
    int n = blockIdx.x * blockDim.x + threadIdx.x;
    if (n >= NGT) return;
    int b = n / T;
    const float* t5 = tgt + n * 5;
    float gx = t5[0] * W, gy = t5[1] * H, gw = t5[2] * W, gh = t5[3] * H;
    int cls = (int)t5[4];
    int gi = (int)floorf(gx); gi = gi < 0 ? 0 : (gi > W - 1 ? W - 1 : gi);
    int gj = (int)floorf(gy); gj = gj < 0 ? 0 : (gj > H - 1 ? H - 1 : gj);

    float bestIou = -1.0f; int best = 0;
    float ious[A];
#pragma unroll
    for (int a = 0; a < A; a++) {
        float aw = anc[2 * a], ah = anc[2 * a + 1];
        float inter = fminf(gw, aw) * fminf(gh, ah);
        float iou = inter / (gw * gh + aw * ah - inter + 1e-16f);
        ious[a] = iou;
        if (iou > bestIou) { bestIou = iou; best = a; }   // first-max like jnp.argmax
    }
    int cellBase = ((b * A) * H + gj) * W + gi;
#pragma unroll
    for (int a = 0; a < A; a++)
        if (ious[a] > IGNORE_THRES) atomicOr(&cellCZ[cellBase + a * HW], 1u);

    int cell = cellBase + best * HW;
    atomicExch(&cellGT[cell], (uint32)(n + 1));   // duplicate scatter: one winner

    // n_correct: argmax over 80 gathered class logits + sigmoid(conf) > 0.5
    const float* row = xc + (size_t)cell * C;
    float mx = row[0]; int lab = 0;
#pragma unroll 4
    for (int c = 1; c < C; c++) { float v = row[c]; if (v > mx) { mx = v; lab = c; } }
    float score = sigmoidf(xr[(size_t)cell * 5 + 4]);
    if (lab == cls && score > 0.5f) atomicAdd((int*)(acc + 9), 1);
}

__global__ void __launch_bounds__(256)
k_dense(const float* __restrict__ xr, const float* __restrict__ xc,
        const float* __restrict__ tgt, const float* __restrict__ anc,
        const uint32* __restrict__ cellGT, const uint32* __restrict__ cellCZ,
        float* acc) {
    __shared__ __align__(16) float s_xr[256 * 5];   // 5120 B tile of x_reg
    __shared__ float s_part[9][8];

    const int tid = threadIdx.x;
    const int base = blockIdx.x * 256;

    // ---- async stage: 5120 B of x_reg -> LDS (gfx1250 async-tensor path) ----
    {
        const char* gsrc = (const char*)xr + (size_t)base * 5 * 4;   // 16B aligned
        unsigned lbase = (unsigned)(uintptr_t)&s_xr[0];
        unsigned loff = lbase + (unsigned)tid * 16u;
        asm volatile("global_load_async_to_lds_b128 %0, %1, off"
                     :: "v"(loff), "v"(gsrc + tid * 16) : "memory");
        if (tid < 64) {
            asm volatile("global_load_async_to_lds_b128 %0, %1, off"
                         :: "v"(loff + 4096u), "v"(gsrc + tid * 16 + 4096) : "memory");
        }
        asm volatile("s_wait_asynccnt 0" ::: "memory");
    }
    __syncthreads();

    const int cell = base + tid;
    float pc = sigmoidf(s_xr[tid * 5 + 4]);
    pc = fminf(fmaxf(pc, EPS_BCE), 1.0f - EPS_BCE);

    uint32 g = cellGT[cell];
    uint32 cz = cellCZ[cell];

    float q[9] = {0.f, 0.f, 0.f, 0.f, 0.f, 0.f, 0.f, 0.f, 0.f};
    if (g) {
        int n = (int)g - 1;
        const float* t5 = tgt + n * 5;
        float gx = t5[0] * W, gy = t5[1] * H, gw = t5[2] * W, gh = t5[3] * H;
        int cls = (int)t5[4];
        int gi = (int)floorf(gx); gi = gi < 0 ? 0 : (gi > W - 1 ? W - 1 : gi);
        int gj = (int)floorf(gy); gj = gj < 0 ? 0 : (gj > H - 1 ? H - 1 : gj);
        int a = (cell / HW) % A;
        float aw = anc[2 * a], ah = anc[2 * a + 1];

        float tx = gx - (float)gi, ty = gy - (float)gj;
        float tw = logf(gw / aw + 1e-16f), th = logf(gh / ah + 1e-16f);

        q[0] = smooth_l1(sigmoidf(s_xr[tid * 5 + 0]), tx);
        q[1] = smooth_l1(sigmoidf(s_xr[tid * 5 + 1]), ty);
        q[2] = smooth_l1(s_xr[tid * 5 + 2], tw);
        q[3] = smooth_l1(s_xr[tid * 5 + 3], th);
        q[4] = -logf(pc);

        // gathered log-softmax CE over 80 classes (only <=512 threads take this)
        const float* row = xc + (size_t)cell * C;
        float mx = -INFINITY;
#pragma unroll 4
        for (int c = 0; c < C; c++) mx = fmaxf(mx, row[c]);
        float s = 0.0f;
#pragma unroll 4
        for (int c = 0; c < C; c++) s += __expf(row[c] - mx);
        q[6] = (mx + logf(s)) - row[cls];
        q[7] = 1.0f;
    } else if (!cz) {
        q[5] = -logf(1.0f - pc);
        q[8] = 1.0f;
    }

    // ---- wave32 shuffle reduction, then WMMA column-sum across waves ----
    const int lane = tid & 31, wv = tid >> 5;
#pragma unroll
    for (int qi = 0; qi < 9; qi++) {
        float v = q[qi];
        for (int m = 16; m > 0; m >>= 1) v += __shfl_xor(v, m, 32);
        if (lane == 0) s_part[qi][wv] = v;
    }
    __syncthreads();

    if (wv == 0) {   // whole wave 0: EXEC all-ones as WMMA requires
#pragma unroll
        for (int qi = 0; qi < 9; qi++) {
            float v = (lane < 8) ? s_part[qi][lane] : 0.0f;
#if __has_builtin(__builtin_amdgcn_wmma_f32_16x16x4_f32)
            // A = all-ones 16x4  =>  D[m,n] = sum_k B[k,n] : layout-agnostic 64->16 reducer
            v2f amat; amat.x = 1.0f; amat.y = 1.0f;
            v2f bmat; bmat.x = v;    bmat.y = 0.0f;
            v8f cmat = {};
            v8f d = __builtin_amdgcn_wmma_f32_16x16x4_f32(
                false, amat, false, bmat, (short)0, cmat, false, false);
            float col = d[0];                 // row M=0 column sums live in lanes 0..15
            col += __shfl_xor(col, 1, 32);
            col += __shfl_xor(col, 2, 32);
            col += __shfl_xor(col, 4, 32);
            col += __shfl_xor(col, 8, 32);
            if (lane == 0) atomicAdd(&acc[qi], col);
#else
            for (int m = 4; m > 0; m >>= 1) v += __shfl_xor(v, m, 32);
            if (lane == 0) atomicAdd(&acc[qi], v);
#endif
        }
    }
}

__global__ void k_final(const float* __restrict__ acc, float* __restrict__ out) {
    if (blockIdx.x == 0 && threadIdx.x == 0) {
        float nM = fmaxf(acc[7], 1.0f);
        float nF = fmaxf(acc[8], 1.0f);
        float lx = acc[0] / nM, ly = acc[1] / nM, lw = acc[2] / nM, lh = acc[3] / nM;
        float lconf = acc[5] / nF + acc[4] / nM;   // NOOBJ_SCALE*noobj + OBJ_SCALE*obj
        float lcls = acc[6] / nM;
        float coord = lx + ly + lw + lh;
        out[0] = coord + lconf + lcls;
        out[1] = coord;
        out[2] = lconf;
        out[3] = lcls;
        out[4] = (float)(*(const int*)(acc + 9));
    }
}

extern "C" void kernel_launch(void* const* d_in, const int* in_sizes, int n_in,
                              void* d_out, int out_size, void* d_ws, size_t ws_size,
                              hipStream_t stream) {
    const float* xr  = (const float*)d_in[0];   // x_reg   [B,A,H,W,5]
    const float* xc  = (const float*)d_in[1];   // x_class [B,A,H,W,C]
    const float* tgt = (const float*)d_in[2];   // target  [B,T,5]
    const float* anc = (const float*)d_in[3];   // anchors [A,2]
    (void)in_sizes; (void)n_in; (void)out_size; (void)ws_size;

    uint32* cellGT = (uint32*)d_ws;
    uint32* cellCZ = cellGT + NCELL;
    float*  acc    = (float*)(cellCZ + NCELL);

    k_init <<<(NCELL + 255) / 256, 256, 0, stream>>>(cellGT, cellCZ, acc);
    k_build<<<(NGT + 255) / 256,   256, 0, stream>>>(xr, xc, tgt, anc, cellGT, cellCZ, acc);
    k_dense<<<NCELL / 256,         256, 0, stream>>>(xr, xc, tgt, anc, cellGT, cellCZ, acc);
    k_final<<<1, 32, 0, stream>>>(acc, (float*)d_out);
}